// MLLA_36137854828668
// MI455X (gfx1250) — compile-verified
//
#include <hip/hip_runtime.h>
#include <math.h>

// MLLA linear-attention block for MI455X (gfx1250, wave32).
// fp32 end-to-end (memory-bound: ~1.3 GB total traffic @ 23.3 TB/s >> 4 GFLOP).
// Matrix einsums (head_dim = 16) use V_WMMA_F32_16X16X4_F32.
// No LDS/TDM staging: Kr/v have zero reuse in the kv reduction, so direct
// coalesced global loads are the roofline-optimal path.
//
// Workspace (floats): Qr(33.5M) | Kr(33.5M) | KV(8192) | KS(131072) | KM(512)
//   total ~269 MB.

#define BB      8
#define CC      64
#define HH      256
#define WW      256
#define NN      65536
#define HEADS   4
#define HD      16
#define NSLICE  256

#define QR_ELEMS (BB * HEADS * NN * HD)   // 33554432
#define KV_ELEMS (BB * HEADS * HD * HD)   // 8192
#define KS_ELEMS (NSLICE * BB * CC)       // 131072
#define KM_ELEMS (BB * CC)                // 512

typedef float v2f __attribute__((ext_vector_type(2)));
typedef float v8f __attribute__((ext_vector_type(8)));

#define LOG1E4_OVER_32 0.2878231366422366f   // ln(10000)/32

__global__ void mlla_zero(float* __restrict__ p, int n) {
  int i = blockIdx.x * blockDim.x + threadIdx.x;
  if (i < n) p[i] = 0.0f;
}

// ---------------------------------------------------------------------------
// Kernel 1: depthwise 3x3 conv (remapped channels) + elu+1 + RoPE.
// Block = 256 threads: t = tid&127 is the conv x-lane for one p-row
// (t<64 -> q channels j=t, t>=64 -> k channels j=t-64), which = tid>>7
// selects one of two p values; 4 iterations -> 8 p rows per block.
// ---------------------------------------------------------------------------
__global__ __launch_bounds__(256) void mlla_prep(
    const float* __restrict__ x, const float* __restrict__ qk_w,
    const float* __restrict__ qk_b, float* __restrict__ Qr,
    float* __restrict__ Kr, float* __restrict__ KS) {
  const int tid   = threadIdx.x;
  const int t     = tid & 127;
  const int which = tid >> 7;
  const int blk   = blockIdx.x;
  const int b     = blk >> 13;                 // NN/8 = 8192 blocks per batch
  const int pbase = (blk & 8191) * 8;
  const int slice = blk & (NSLICE - 1);

  const int  j    = t & 63;
  const bool is_k = (t >= 64);
  const int  head = j >> 4;
  const int  d    = j & 15;

  const float theta = __expf(-(float)(j >> 1) * LOG1E4_OVER_32);

#pragma unroll
  for (int it = 0; it < 4; ++it) {
    const int p   = pbase + 2 * it + which;
    const int hi  = p >> 9;                    // conv output channel 0..127
    const int mid = (p >> 1) & 255;            // y
    const int lo  = p & 1;
    const int xx  = lo * 128 + t;              // conv x coordinate 0..255
    const int ic  = hi >> 1;                   // input channel (groups=64)

    const float* __restrict__ xc = x + (((size_t)b * CC + ic) << 16);
    const float* __restrict__ wv = qk_w + hi * 9;
    float sum = qk_b[hi];
#pragma unroll
    for (int dy = -1; dy <= 1; ++dy) {
      const int yy = mid + dy;
      if (yy < 0 || yy >= HH) continue;
      const float* row = xc + (yy << 8);
#pragma unroll
      for (int dx = -1; dx <= 1; ++dx) {
        const int xp = xx + dx;
        if (xp < 0 || xp >= WW) continue;
        sum = fmaf(row[xp], wv[(dy + 1) * 3 + (dx + 1)], sum);
      }
    }
    // elu(x)+1 = x+1 (x>0) else exp(x)
    const float val = (sum > 0.0f) ? (sum + 1.0f) : __expf(sum);

    if (is_k)  // non-roped k feeds k_mean (sliced to spread atomic contention)
      atomicAdd(&KS[slice * (BB * CC) + b * CC + j], val);

    // RoPE: pairs (2kk, 2kk+1) are adjacent lanes within the wave32
    float s, c;
    __sincosf((float)(p & 255) * theta, &s, &c);
    const float partner = __shfl_xor(val, 1, 32);
    const float roped = ((j & 1) == 0) ? fmaf(val, c, -partner * s)
                                       : fmaf(partner, s, val * c);

    const int idx = (((b * HEADS + head) * NN) + p) * HD + d;
    if (is_k) Kr[idx] = roped;
    else      Qr[idx] = roped;
  }
}

// ---------------------------------------------------------------------------
// Kernel 2: kv[bh] = (1/N) * sum_p Kr[p,:]^T v[p,:]  via WMMA f32 16x16x4.
// 128 waves per (b,head), 512 positions per wave, partials combined with
// global_atomic_add_f32.
// Constant 128-trip loop (SALU loop control, no exec masking) + unroll 4
// -> 4 WMMAs per body with loads folded into immediate offsets.
// A(16x4): lanes 0-15 hold (M=d, K=0/1), lanes 16-31 hold (M=d, K=2/3).
// B(4x16): lanes 0-15 hold (K=0/1, N=e), lanes 16-31 hold (K=2/3, N=e).
// ---------------------------------------------------------------------------
__global__ __launch_bounds__(256) void mlla_kv(
    const float* __restrict__ x, const float* __restrict__ Kr,
    float* __restrict__ KV) {
  const int wave  = threadIdx.x >> 5;
  const int lane  = threadIdx.x & 31;
  const int bh    = blockIdx.x >> 4;              // 0..31
  const int wslot = (blockIdx.x & 15) * 8 + wave; // 0..127
  const int b     = bh >> 2;
  const int head  = bh & 3;
  const int half  = lane >> 4;
  const int l     = lane & 15;

  const int pbeg = wslot * 512 + 2 * half;
  const float* __restrict__ kp =
      Kr + (size_t)bh * NN * HD + (size_t)pbeg * HD + l;
  const float* __restrict__ vp =
      x + (((size_t)b * CC) << 16) + (size_t)pbeg * CC + head * HD + l;

  v8f acc = {0.f, 0.f, 0.f, 0.f, 0.f, 0.f, 0.f, 0.f};
#pragma unroll 4
  for (int i = 0; i < 128; ++i) {
    if ((i & 3) == 0)  // resolved per unrolled copy: 1 prefetch / 16 positions
      __builtin_prefetch(kp + 32 * HD, 0, 1);
    v2f a, bv;
    a.x  = kp[0];      // coalesced: 16 lanes x 4B contiguous per half-wave
    a.y  = kp[HD];
    bv.x = vp[0];      // v rows (16 floats / head) contiguous in x
    bv.y = vp[CC];
    acc  = __builtin_amdgcn_wmma_f32_16x16x4_f32(false, a, false, bv,
                                                 (short)0, acc, false, false);
    kp += 4 * HD;
    vp += 4 * CC;
  }
  float* kvt = KV + bh * (HD * HD);
  const float scale = 1.0f / 65536.0f;           // (n^-1/2)^2
#pragma unroll
  for (int r = 0; r < 8; ++r)
    atomicAdd(&kvt[(r + half * 8) * HD + l], acc[r] * scale);
}

// ---------------------------------------------------------------------------
// Kernel 3: reduce k-sum slices -> k_mean
// ---------------------------------------------------------------------------
__global__ void mlla_kmean(const float* __restrict__ KS,
                           float* __restrict__ KM) {
  const int i = blockIdx.x * blockDim.x + threadIdx.x;
  if (i >= BB * CC) return;
  float s = 0.0f;
  for (int sl = 0; sl < NSLICE; ++sl) s += KS[sl * (BB * CC) + i];
  KM[i] = s * (1.0f / 65536.0f);
}

// ---------------------------------------------------------------------------
// Kernel 4: out = z * (Q_rope @ kv) + lepe(v_img), fused + transposed store.
// One 16-position x 16-channel tile per wave-iteration, 4 WMMAs (K=16).
// z = 1/(q.k_mean+eps) computed from Q_rope via z-dot = Qr . R(pw) k_mean.
// ---------------------------------------------------------------------------
__global__ __launch_bounds__(256) void mlla_out(
    const float* __restrict__ x, const float* __restrict__ Qr,
    const float* __restrict__ KV, const float* __restrict__ KM,
    const float* __restrict__ lepe_w, const float* __restrict__ lepe_b,
    float* __restrict__ out) {
  const int wave = threadIdx.x >> 5;
  const int lane = threadIdx.x & 31;
  const int half = lane >> 4;
  const int l    = lane & 15;
  const int gw   = blockIdx.x * 8 + wave;  // 0..16383
  const int bh   = gw >> 9;
  const int wt   = gw & 511;
  const int b    = bh >> 2;
  const int head = bh & 3;
  const int j    = head * HD + l;

  const float* __restrict__ Qrb = Qr + (size_t)bh * NN * HD;
  const float* __restrict__ xb  = x + (((size_t)b * CC) << 16);
  const float* __restrict__ kvt = KV + bh * (HD * HD);

  // kv as 4 B-matrices (K chunks of 4), loaded once per wave
  v2f Bm[4];
#pragma unroll
  for (int tt = 0; tt < 4; ++tt) {
    const int d0 = 4 * tt + 2 * half;
    Bm[tt].x = kvt[d0 * HD + l];
    Bm[tt].y = kvt[(d0 + 1) * HD + l];
  }
  float lw[9];
#pragma unroll
  for (int k9 = 0; k9 < 9; ++k9) lw[k9] = lepe_w[j * 9 + k9];
  const float lb = lepe_b[j];
  float km[16];
#pragma unroll
  for (int dd = 0; dd < 16; ++dd) km[dd] = KM[b * CC + head * HD + dd];
  float th8[8];
#pragma unroll
  for (int kk = 0; kk < 8; ++kk)
    th8[kk] = __expf(-(float)(head * 8 + kk) * LOG1E4_OVER_32);

  for (int ti = 0; ti < 8; ++ti) {
    const int p0 = (wt * 8 + ti) * 16;

    // z for row m = l (lanes 16-31 duplicate), then broadcast via shfl
    const int pz = p0 + l;
    const float* qrow = Qrb + (size_t)pz * HD;
    const int pw = pz & 255;
    float dot = 0.0f;
#pragma unroll
    for (int kk = 0; kk < 8; ++kk) {
      float s, c;
      __sincosf((float)pw * th8[kk], &s, &c);
      const float rre = fmaf(km[2 * kk], c, -km[2 * kk + 1] * s);
      const float rim = fmaf(km[2 * kk], s, km[2 * kk + 1] * c);
      dot = fmaf(qrow[2 * kk], rre, dot);
      dot = fmaf(qrow[2 * kk + 1], rim, dot);
    }
    const float zown = 1.0f / (dot + 1e-6f);

    v8f acc = {0.f, 0.f, 0.f, 0.f, 0.f, 0.f, 0.f, 0.f};
#pragma unroll
    for (int tt = 0; tt < 4; ++tt) {
      v2f a;
      const float* qa = Qrb + (size_t)(p0 + l) * HD + 4 * tt + 2 * half;
      a.x = qa[0];
      a.y = qa[1];
      acc = __builtin_amdgcn_wmma_f32_16x16x4_f32(false, a, false, Bm[tt],
                                                  (short)0, acc, false, false);
    }

    // z-scale + fused lepe 3x3 + transposed store (8 consecutive p per lane)
    const int y = p0 >> 8;
    float o[8];
#pragma unroll
    for (int r = 0; r < 8; ++r) {
      const int M = r + half * 8;
      const int p = p0 + M;
      const float zr = __shfl(zown, M, 32);
      const int xcoord = p & 255;
      float lep = lb;
#pragma unroll
      for (int dy = -1; dy <= 1; ++dy) {
        const int yy = y + dy;
        if (yy < 0 || yy >= HH) continue;
#pragma unroll
        for (int dx = -1; dx <= 1; ++dx) {
          const int xp = xcoord + dx;
          if (xp < 0 || xp >= WW) continue;
          lep = fmaf(xb[(size_t)(((yy << 8) + xp) << 6) + j],
                     lw[(dy + 1) * 3 + (dx + 1)], lep);
        }
      }
      o[r] = fmaf(acc[r], zr, lep);
    }
    float* op = out + (((size_t)(b * CC + j)) << 16) + p0 + half * 8;
    *(float4*)op       = make_float4(o[0], o[1], o[2], o[3]);
    *(float4*)(op + 4) = make_float4(o[4], o[5], o[6], o[7]);
  }
}

extern "C" void kernel_launch(void* const* d_in, const int* in_sizes, int n_in,
                              void* d_out, int out_size, void* d_ws,
                              size_t ws_size, hipStream_t stream) {
  const float* x      = (const float*)d_in[0];
  const float* qk_w   = (const float*)d_in[1];
  const float* qk_b   = (const float*)d_in[2];
  const float* lepe_w = (const float*)d_in[3];
  const float* lepe_b = (const float*)d_in[4];
  float* out = (float*)d_out;

  float* W  = (float*)d_ws;
  float* Qr = W;
  float* Kr = Qr + QR_ELEMS;
  float* KV = Kr + QR_ELEMS;
  float* KS = KV + KV_ELEMS;
  float* KM = KS + KS_ELEMS;

  const int zn = KV_ELEMS + KS_ELEMS + KM_ELEMS;
  mlla_zero<<<(zn + 255) / 256, 256, 0, stream>>>(KV, zn);
  mlla_prep<<<BB * NN / 8, 256, 0, stream>>>(x, qk_w, qk_b, Qr, Kr, KS);
  mlla_kv<<<32 * 16, 256, 0, stream>>>(x, Kr, KV);
  mlla_kmean<<<2, 256, 0, stream>>>(KS, KM);
  mlla_out<<<2048, 256, 0, stream>>>(x, Qr, KV, KM, lepe_w, lepe_b, out);
}